// DynamicDownsampling_65343632441723
// MI455X (gfx1250) — compile-verified
//
#include <hip/hip_runtime.h>
#include <cstdint>

// x[2,3,3,640,640], kernel[2,16,3,320,320], DT[2,1,3,640,640] -> out[2,3,320,320]
// K=4, S=2, PAD=1 (edge replicate). Memory-bound: ~81 MB @ 23.3 TB/s => ~3.5 us floor.
#define IW   640
#define IH   640
#define IHW  (IW * IH)          // 409600
#define OW   320
#define OH   320
#define OHW  (OW * OH)          // 102400

#define TBW  32                 // output tile width  (threads x)
#define TBH  8                  // output tile height (threads y)
#define TILE_C  66              // input tile cols = TBW*2 + 2
#define TILE_R  18              // input tile rows = TBH*2 + 2
#define TILE_CP 68              // col pitch: stride-2 reads hit 32 distinct banks of 64
#define PLANE   (TILE_R * TILE_CP)    // 1224 words
#define PLANE_B (PLANE * 4)           // 4896 bytes
#define NPLANES 12              // (3 x-channels + DT) * 3 t

__device__ __forceinline__ int clampi(int v, int lo, int hi) {
    return v < lo ? lo : (v > hi ? hi : v);
}

__global__ __launch_bounds__(TBW * TBH)
void dyndown_kernel(const float* __restrict__ x,
                    const float* __restrict__ kern,
                    const float* __restrict__ dt,
                    float* __restrict__ out)
{
    __shared__ float smem[NPLANES * PLANE];   // 58,752 bytes

    const int b  = blockIdx.z;
    const int w0 = blockIdx.x * TBW;
    const int h0 = blockIdx.y * TBH;
    const int tx = threadIdx.x;   // 0..31
    const int ty = threadIdx.y;   // 0..7 (wave-uniform on wave32)

    // Low 32 bits of a generic pointer to LDS == byte offset in the WG's LDS
    // allocation (what async-to-LDS VDST expects).
    const unsigned lds_base = (unsigned)(uintptr_t)&smem[0];

    // ---- Hoisted copy coordinates (reused across all 12 planes) -------------
    // Dense 2x2: rows {ty, ty+8} x cols {tx, tx+32}  == rows 0..15 x cols 0..63
    unsigned gbD[2][2], lbD[2][2];
    #pragma unroll
    for (int qr = 0; qr < 2; ++qr) {
        int rr = ty + 8 * qr;
        int gr = clampi(2 * h0 - 1 + rr, 0, IH - 1);
        #pragma unroll
        for (int qc = 0; qc < 2; ++qc) {
            int cc = tx + 32 * qc;
            int gc = clampi(2 * w0 - 1 + cc, 0, IW - 1);
            gbD[qr][qc] = (unsigned)(gr * IW + gc) * 4u;
            lbD[qr][qc] = lds_base + (unsigned)(rr * TILE_CP + cc) * 4u;
        }
    }
    // Edge strips, one (uniform) assignment per wave:
    //  ty==0: cols 64..65 x rows 0..15   (1 load/plane)
    //  ty==1: row 16 x cols 0..63        (2 loads/plane)
    //  ty==2: row 17 x cols 0..63        (2 loads/plane)
    //  ty==3: rows 16..17 x cols 64..65  (1 load/plane; lanes 4..31 duplicate
    //         lanes 0..3 -> identical data to identical LDS address: benign)
    unsigned gbE[2], lbE[2];
    int nextra = 0;
    {
        int r0 = 0, c0 = 0, r1 = 0, c1 = 0;
        if (ty == 0)      { r0 = tx >> 1;             c0 = 64 + (tx & 1); nextra = 1; }
        else if (ty == 1) { r0 = 16; c0 = tx; r1 = 16; c1 = tx + 32;      nextra = 2; }
        else if (ty == 2) { r0 = 17; c0 = tx; r1 = 17; c1 = tx + 32;      nextra = 2; }
        else if (ty == 3) { r0 = 16 + ((tx >> 1) & 1); c0 = 64 + (tx & 1); nextra = 1; }
        int g0r = clampi(2 * h0 - 1 + r0, 0, IH - 1);
        int g0c = clampi(2 * w0 - 1 + c0, 0, IW - 1);
        int g1r = clampi(2 * h0 - 1 + r1, 0, IH - 1);
        int g1c = clampi(2 * w0 - 1 + c1, 0, IW - 1);
        gbE[0] = (unsigned)(g0r * IW + g0c) * 4u;
        lbE[0] = lds_base + (unsigned)(r0 * TILE_CP + c0) * 4u;
        gbE[1] = (unsigned)(g1r * IW + g1c) * 4u;
        lbE[1] = lds_base + (unsigned)(r1 * TILE_CP + c1) * 4u;
    }

    // ---- Phase 1: async-copy 12 planes into LDS (ASYNCcnt-tracked) ---------
    // GVS form: uniform SGPR-pair plane base + per-lane 32-bit byte offset.
    #pragma unroll
    for (int a = 0; a < 4; ++a) {
        #pragma unroll
        for (int t = 0; t < 3; ++t) {
            const float* src = (a < 3)
                ? (x  + (size_t)((b * 3 + a) * 3 + t) * IHW)
                : (dt + (size_t)(b * 3 + t) * IHW);
            const unsigned pb = (unsigned)((a * 3 + t) * PLANE_B);
            #pragma unroll
            for (int qr = 0; qr < 2; ++qr)
                #pragma unroll
                for (int qc = 0; qc < 2; ++qc) {
                    unsigned laddr = lbD[qr][qc] + pb;
                    asm volatile("global_load_async_to_lds_b32 %0, %1, %2"
                                 :: "v"(laddr), "v"(gbD[qr][qc]), "s"(src)
                                 : "memory");
                }
            for (int e = 0; e < nextra; ++e) {           // wave-uniform count
                unsigned laddr = lbE[e] + pb;
                asm volatile("global_load_async_to_lds_b32 %0, %1, %2"
                             :: "v"(laddr), "v"(gbE[e]), "s"(src)
                             : "memory");
            }
        }
    }

    // ---- Phase 2: per-pixel softmax logits (overlaps async copies) ---------
    const int hw = (h0 + ty) * OW + (w0 + tx);
    const float* kbase = kern + (size_t)b * 16 * 3 * OHW + hw;

    float lg[48];                            // flat softmax order: t*16 + kk
    #pragma unroll
    for (int kk = 0; kk < 16; ++kk)
        #pragma unroll
        for (int t = 0; t < 3; ++t)
            lg[t * 16 + kk] = kbase[(size_t)(kk * 3 + t) * OHW];

    float m = lg[0];
    #pragma unroll
    for (int i = 1; i < 48; ++i) m = fmaxf(m, lg[i]);
    float s = 0.f;
    #pragma unroll
    for (int i = 0; i < 48; ++i) { lg[i] = __expf(lg[i] - m); s += lg[i]; }

    // Drain this wave's async copies, then make them visible workgroup-wide.
    asm volatile("s_wait_asynccnt 0" ::: "memory");
    __syncthreads();

    // ---- Phase 3: windowed weighted accumulation from LDS ------------------
    float ax0 = 0.f, ax1 = 0.f, ax2 = 0.f, ad = 0.f;
    #pragma unroll
    for (int t = 0; t < 3; ++t) {
        #pragma unroll
        for (int i = 0; i < 4; ++i) {
            #pragma unroll
            for (int j = 0; j < 4; ++j) {
                float wgt = lg[t * 16 + i * 4 + j];
                int o = (2 * ty + i) * TILE_CP + (2 * tx + j);
                ax0 += wgt * smem[(0 * 3 + t) * PLANE + o];
                ax1 += wgt * smem[(1 * 3 + t) * PLANE + o];
                ax2 += wgt * smem[(2 * 3 + t) * PLANE + o];
                ad  += wgt * smem[(3 * 3 + t) * PLANE + o];
            }
        }
    }

    // out = (ax/s) / (ad/s + 1e-8) == ax / (ad + 1e-8*s)
    float denom = ad + 1e-8f * s;
    float r0 = ax0 / denom, r1 = ax1 / denom, r2 = ax2 / denom;

    size_t ob = (size_t)(b * 3) * OHW + hw;
    out[ob]           = r0;
    out[ob + OHW]     = r1;
    out[ob + 2 * OHW] = r2;
}

extern "C" void kernel_launch(void* const* d_in, const int* in_sizes, int n_in,
                              void* d_out, int out_size, void* d_ws, size_t ws_size,
                              hipStream_t stream) {
    const float* x  = (const float*)d_in[0];   // [2,3,3,640,640]
    const float* k  = (const float*)d_in[1];   // [2,16,3,320,320]
    const float* dt = (const float*)d_in[2];   // [2,1,3,640,640]
    float* outp = (float*)d_out;               // [2,3,320,320]
    dim3 block(TBW, TBH, 1);
    dim3 grid(OW / TBW, OH / TBH, 2);
    hipLaunchKernelGGL(dyndown_kernel, grid, block, 0, stream, x, k, dt, outp);
}